// GaussianBasisFunctions_87376814670609
// MI455X (gfx1250) — compile-verified
//
#include <hip/hip_runtime.h>
#include <hip/hip_bf16.h>
#include <math.h>

typedef __attribute__((ext_vector_type(2))) float v2f;
typedef __attribute__((ext_vector_type(4))) float v4f;
typedef __attribute__((ext_vector_type(8))) float v8f;

#define MPTS 65536
#define NGS  2048
#define KCO  16          // coefficient row stride (12 used, padded)
#define NSUP 512         // columns staged in LDS per super-chunk (32 KB)
#define EPS_QUAT 1e-8f
#define EPS_COV  1e-6f

// -0.5 * log2(e): folded into the coefficients so WMMA directly yields the
// exp2 argument and the epilogue needs no per-element scaling multiply.
#define NEG_HALF_L2E (-0.72134752044448170368f)

// -------- Kernel 1: per-Gaussian coefficient vector (N=2048 threads) --------
__global__ void gbf_prep(const float* __restrict__ pos,
                         const float* __restrict__ lsc,
                         const float* __restrict__ rot,
                         const float* __restrict__ wts,
                         float* __restrict__ coef) {
    int n = blockIdx.x * blockDim.x + threadIdx.x;
    if (n >= NGS) return;

    float ux = pos[n*3+0], uy = pos[n*3+1], uz = pos[n*3+2];
    float e0 = __expf(lsc[n*3+0]);
    float e1 = __expf(lsc[n*3+1]);
    float e2 = __expf(lsc[n*3+2]);
    float s0 = e0*e0, s1 = e1*e1, s2 = e2*e2;

    float qw = rot[n*4+0], qx = rot[n*4+1], qy = rot[n*4+2], qz = rot[n*4+3];
    float inv = 1.0f / (sqrtf(qw*qw + qx*qx + qy*qy + qz*qz) + EPS_QUAT);
    qw *= inv; qx *= inv; qy *= inv; qz *= inv;

    float r00 = 1.f - 2.f*(qy*qy + qz*qz);
    float r01 = 2.f*(qx*qy - qz*qw);
    float r02 = 2.f*(qx*qz + qy*qw);
    float r10 = 2.f*(qx*qy + qz*qw);
    float r11 = 1.f - 2.f*(qx*qx + qz*qz);
    float r12 = 2.f*(qy*qz - qx*qw);
    float r20 = 2.f*(qx*qz - qy*qw);
    float r21 = 2.f*(qy*qz + qx*qw);
    float r22 = 1.f - 2.f*(qx*qx + qy*qy);

    // cov = R diag(s^2) R^T + eps I  (symmetric)
    float c00 = r00*r00*s0 + r01*r01*s1 + r02*r02*s2 + EPS_COV;
    float c01 = r00*r10*s0 + r01*r11*s1 + r02*r12*s2;
    float c02 = r00*r20*s0 + r01*r21*s1 + r02*r22*s2;
    float c11 = r10*r10*s0 + r11*r11*s1 + r12*r12*s2 + EPS_COV;
    float c12 = r10*r20*s0 + r11*r21*s1 + r12*r22*s2;
    float c22 = r20*r20*s0 + r21*r21*s1 + r22*r22*s2 + EPS_COV;

    // symmetric 3x3 inverse via adjugate
    float m00 = c11*c22 - c12*c12;
    float m01 = c02*c12 - c01*c22;
    float m02 = c01*c12 - c02*c11;
    float det = c00*m00 + c01*m01 + c02*m02;
    float id  = 1.0f / det;
    float A00 = m00*id;
    float A01 = m01*id;
    float A02 = m02*id;
    float A11 = (c00*c22 - c02*c02)*id;
    float A12 = (c01*c02 - c00*c12)*id;
    float A22 = (c00*c11 - c01*c01)*id;

    float b0 = A00*ux + A01*uy + A02*uz;
    float b1 = A01*ux + A11*uy + A12*uz;
    float b2 = A02*ux + A12*uy + A22*uz;
    float uu = ux*b0 + uy*b1 + uz*b2;

    float w   = wts[n];
    float l2w = __log2f(fmaxf(w, 1e-30f));  // fold weight into quad constant

    // Pre-scale by -0.5*log2(e): feature dot product == exp2 argument.
    const float S = NEG_HALF_L2E;
    float* c = coef + n*KCO;
    c[0]  = S*A00;        c[1]  = S*A11;        c[2]  = S*A22;
    c[3]  = S*2.f*A01;    c[4]  = S*2.f*A02;    c[5]  = S*2.f*A12;
    c[6]  = S*-2.f*b0;    c[7]  = S*-2.f*b1;    c[8]  = S*-2.f*b2;
    c[9]  = S*uu + l2w;
    c[10] = 0.f; c[11] = 0.f; c[12] = 0.f; c[13] = 0.f; c[14] = 0.f; c[15] = 0.f;
}

// -------- Kernel 2: M x N quad GEMM via V_WMMA_F32_16X16X4_F32 + exp epilogue --------
__global__ __launch_bounds__(256) void gbf_main(const float* __restrict__ pts,
                                                const float* __restrict__ coef,
                                                float* __restrict__ out) {
    __shared__ float scoef[NSUP * KCO];   // 32 KB coefficient tile

    const int tid  = threadIdx.x;
    const int lane = tid & 31;
    const int wave = tid >> 5;
    const int hi   = lane >> 4;           // which K-interleave half this lane supplies
    const int mrow = lane & 15;
    const int rowBase = blockIdx.x * (8 * 16) + wave * 16;

    // Per-lane loop-invariant A operand (features of point rowBase+mrow).
    // A-matrix 16x4 f32 layout: VGPR0 = K=(4kk+0 | 4kk+2), VGPR1 = K=(4kk+1 | 4kk+3).
    const int prow = rowBase + mrow;
    float x = pts[prow*3+0], y = pts[prow*3+1], z = pts[prow*3+2];
    v2f a0, a1, a2;
    if (hi) { a0 = (v2f){z*z, x*y}; a1 = (v2f){x, y};     a2 = (v2f){0.f, 0.f}; }
    else    { a0 = (v2f){x*x, y*y}; a1 = (v2f){x*z, y*z}; a2 = (v2f){z, 1.f};   }

    float partial[8];
#pragma unroll
    for (int r = 0; r < 8; ++r) partial[r] = 0.f;

    for (int ns = 0; ns < NGS; ns += NSUP) {
        __syncthreads();
        // cooperative stage: NSUP*KCO floats = 2048 float4, 8 per thread
        {
            const v4f* src = (const v4f*)(coef + ns * KCO);
            v4f* dst = (v4f*)scoef;
#pragma unroll
            for (int i = 0; i < (NSUP*KCO/4)/256; ++i)
                dst[tid + i*256] = src[tid + i*256];
        }
        __syncthreads();

        for (int cc = 0; cc < NSUP; cc += 16) {
            // B-matrix 4x16 f32: VGPR0 = K=(4kk+0 | 4kk+2), VGPR1 = K=(4kk+1 | 4kk+3);
            // this lane supplies column (cc+mrow), K pair (k0, k0+1), k0 = 4kk + 2*hi.
            const float* bc = scoef + (cc + mrow) * KCO + (hi ? 2 : 0);
            v2f b0 = (v2f){bc[0], bc[1]};   // K 0..3
            v2f b1 = (v2f){bc[4], bc[5]};   // K 4..7
            v2f b2 = (v2f){bc[8], bc[9]};   // K 8..11

            v8f acc = {};
            acc = __builtin_amdgcn_wmma_f32_16x16x4_f32(false, a0, false, b0,
                                                        (short)0, acc, false, false);
            acc = __builtin_amdgcn_wmma_f32_16x16x4_f32(false, a1, false, b1,
                                                        (short)0, acc, false, false);
            acc = __builtin_amdgcn_wmma_f32_16x16x4_f32(false, a2, false, b2,
                                                        (short)0, acc, false, false);

            // acc already holds log2 of the Gaussian contribution (scale folded
            // into coefficients): epilogue is pure v_exp_f32 + v_pk_add_f32.
#pragma unroll
            for (int r = 0; r < 8; ++r)
                partial[r] += __builtin_amdgcn_exp2f(acc[r]);
        }
    }

    // Reduce the 16 columns held across lanes {0..15} / {16..31}.
    // C/D layout: VGPR r -> row (rowBase + r) for lanes 0-15, (rowBase + 8 + r) for 16-31.
#pragma unroll
    for (int r = 0; r < 8; ++r) {
        float v = partial[r];
        v += __shfl_xor(v, 1, 32);
        v += __shfl_xor(v, 2, 32);
        v += __shfl_xor(v, 4, 32);
        v += __shfl_xor(v, 8, 32);
        if (mrow == 0) out[rowBase + hi*8 + r] = v;
    }
}

extern "C" void kernel_launch(void* const* d_in, const int* in_sizes, int n_in,
                              void* d_out, int out_size, void* d_ws, size_t ws_size,
                              hipStream_t stream) {
    const float* points    = (const float*)d_in[0];  // M*3
    const float* positions = (const float*)d_in[1];  // N*3
    const float* log_scales= (const float*)d_in[2];  // N*3
    const float* rotations = (const float*)d_in[3];  // N*4
    const float* weights   = (const float*)d_in[4];  // N
    float* out  = (float*)d_out;                     // M
    float* coef = (float*)d_ws;                      // N*KCO floats = 128 KB

    gbf_prep<<<(NGS + 255)/256, 256, 0, stream>>>(positions, log_scales, rotations,
                                                  weights, coef);
    gbf_main<<<MPTS / (8*16), 256, 0, stream>>>(points, coef, out);
}